// SpatialAwareFocalLoss_65274912965219
// MI455X (gfx1250) — compile-verified
//
#include <hip/hip_runtime.h>
#include <math.h>

#define SEQ_LEN   1024
#define NCLS      8
#define ALPHA     0.01f
#define SP_WEIGHT 0.3f
#define LINE_WIN  2
#define BLOCK_T   256   // threads per block; 4 blocks cover one sequence's outputs

#if defined(__gfx1250__) && __has_builtin(__builtin_amdgcn_global_load_async_to_lds_b128)
#define HAVE_ASYNC_LDS 1
#else
#define HAVE_ASYNC_LDS 0
#endif

#if __has_builtin(__builtin_amdgcn_rcpf)
#define FAST_RCP(x) __builtin_amdgcn_rcpf(x)   // v_rcp_f32 (TRANS pipe)
#else
#define FAST_RCP(x) (1.0f / (x))
#endif

// builtin signature (from hipcc diagnostic): pointers to 16B int vectors,
// global (AS1) source, LDS (AS3) destination, imm offset, imm cpol.
typedef int v4i_vs __attribute__((vector_size(16)));
typedef __attribute__((address_space(1))) v4i_vs glb_v4i;
typedef __attribute__((address_space(3))) v4i_vs lds_v4i;

__global__ void safl_zero_kernel(float* out) { out[0] = 0.0f; }

__global__ __launch_bounds__(BLOCK_T) void safl_kernel(
    const float* __restrict__ pred,
    const float* __restrict__ target,
    const int*   __restrict__ lines,
    float* __restrict__ out,
    float inv_total)
{
    __shared__ int    s_lines[SEQ_LEN];     // 4 KB
    __shared__ float2 s_wt[SEQ_LEN];        // 8 KB: {sigmoid row-sum, target row-sum}
    __shared__ float  s_red[BLOCK_T / 32];

    const int t       = threadIdx.x;        // 0..255
    const int q       = blockIdx.x & 3;     // quarter of the sequence we own
    const int b       = blockIdx.x >> 2;    // sequence index
    const int seqBase = b * SEQ_LEN;

    // ---- Stage line indices straight into LDS (CDNA5 async global->LDS path) ----
#if HAVE_ASYNC_LDS
    {
        // each thread copies 16B: lines[seqBase + 4t .. 4t+3] -> s_lines[4t..]
        glb_v4i* gp = (glb_v4i*)(lines + seqBase + t * 4);
        lds_v4i* lp = (lds_v4i*)&s_lines[t * 4];
        __builtin_amdgcn_global_load_async_to_lds_b128(gp, lp, 0, 0);
#if __has_builtin(__builtin_amdgcn_s_wait_asynccnt)
        __builtin_amdgcn_s_wait_asynccnt(0);
#else
        asm volatile("s_wait_asynccnt 0x0" ::: "memory");
#endif
    }
#else
    {
        *(int4*)&s_lines[t * 4] = *(const int4*)(lines + seqBase + t * 4);
    }
#endif

    // ---- Stage {sigmoid-row-sum, target-row-sum} for the WHOLE sequence; the
    //      penalty only ever needs class-summed quantities (it feeds a global
    //      mean), so per-class sigmoids never touch LDS. Focal loss is computed
    //      only for the quarter this block owns.
    //
    //      All transcendentals go through the TRANS pipe via closed forms built
    //      on u = exp(-|x|):
    //        sigmoid(x)       = (x>=0 ? 1 : u) / (1+u)
    //        bce              = max(x,0) - x*t + log(1+u)
    //        pt = exp(-bce)   = (t ? (x>=0?1:u) : (x>=0?u:1)) / (1+u)
    //      -> one v_exp + one v_rcp per staged element, one extra v_log per
    //         owned element; no second exp for pt. ----
    float mySigRow = 0.0f; // own token's sigmoid row-sum (for self-exclusion)
    float myTRow   = 0.0f; // own token's target row-sum  (for self-exclusion)
    float focalSum = 0.0f; // own token's focal terms summed over 8 classes

    #pragma unroll
    for (int chunk = 0; chunk < 4; ++chunk) {
        const int tok = chunk * BLOCK_T + t;          // 0..1023
        const size_t row = (size_t)(seqBase + tok) * NCLS;
        const float4 p0 = ((const float4*)(pred   + row))[0];
        const float4 p1 = ((const float4*)(pred   + row))[1];
        const float4 t0 = ((const float4*)(target + row))[0];
        const float4 t1 = ((const float4*)(target + row))[1];

        float x[NCLS]  = {p0.x, p0.y, p0.z, p0.w, p1.x, p1.y, p1.z, p1.w};
        float tv[NCLS] = {t0.x, t0.y, t0.z, t0.w, t1.x, t1.y, t1.z, t1.w};

        float sigSum = 0.0f, tSum = 0.0f;
        float u[NCLS], inv1pu[NCLS];
        #pragma unroll
        for (int c = 0; c < NCLS; ++c) {
            u[c]      = __expf(-fabsf(x[c]));          // v_exp_f32
            inv1pu[c] = FAST_RCP(1.0f + u[c]);         // v_rcp_f32
            const float num = (x[c] >= 0.0f) ? 1.0f : u[c];
            sigSum += num * inv1pu[c];                 // sigmoid(x)
            tSum   += tv[c];
        }
        s_wt[tok] = make_float2(sigSum, tSum);

        if (chunk == q) {
            #pragma unroll
            for (int c = 0; c < NCLS; ++c) {
                const float xv = x[c], tvv = tv[c];
                // numerically stable BCE with logits
                const float bce = fmaxf(xv, 0.0f) - xv * tvv
                                + __logf(1.0f + u[c]); // v_log_f32
                // pt = exp(-bce), closed form sharing u and 1/(1+u)
                const float num = (tvv > 0.5f) ? ((xv >= 0.0f) ? 1.0f : u[c])
                                               : ((xv >= 0.0f) ? u[c] : 1.0f);
                const float pt  = num * inv1pu[c];
                const float om  = 1.0f - pt;
                const float om2 = om * om;
                focalSum += ALPHA * om2 * om2 * bce;
            }
            mySigRow = sigSum;
            myTRow   = tSum;
        }
    }
    __syncthreads();

    // ---- Band window via binary search (lines sorted within sequence) ----
    const int i      = q * BLOCK_T + t;   // own token within sequence
    const int myline = s_lines[i];

    int lo = 0, hi = SEQ_LEN;
    const int vlo = myline - LINE_WIN;
    while (lo < hi) { int m = (lo + hi) >> 1; if (s_lines[m] < vlo) lo = m + 1; else hi = m; }
    const int wlo = lo;

    lo = 0; hi = SEQ_LEN;
    const int vhi = myline + LINE_WIN;
    while (lo < hi) { int m = (lo + hi) >> 1; if (s_lines[m] <= vhi) lo = m + 1; else hi = m; }
    const int whi = lo - 1;               // inclusive; window always contains i

    // ---- Banded "einsum": two scalar windowed sums per token ----
    float accW = 0.0f, accT = 0.0f;
    for (int j = wlo; j <= whi; ++j) {
        const float2 v = s_wt[j];         // single ds_load_b64
        accW += v.x;
        accT += v.y;
    }
    accW -= mySigRow;                     // exclude self (diagonal masked out)
    accT -= myTRow;
    const int counts = whi - wlo;         // window size minus self

    float threadTotal = focalSum;
    if (counts > 0 && accT > 0.0f) {
        threadTotal += (SP_WEIGHT * 0.1f / (float)counts) * accW;
    }

    // ---- wave32 shuffle reduction, then one atomic per block ----
    float v = threadTotal;
    #pragma unroll
    for (int off = 16; off > 0; off >>= 1) v += __shfl_down(v, off, 32);
    const int lane = t & 31, wid = t >> 5;
    if (lane == 0) s_red[wid] = v;
    __syncthreads();
    if (wid == 0) {
        float r = (lane < (BLOCK_T / 32)) ? s_red[lane] : 0.0f;
        #pragma unroll
        for (int off = 16; off > 0; off >>= 1) r += __shfl_down(r, off, 32);
        if (lane == 0) atomicAdd(out, r * inv_total);
    }
}

extern "C" void kernel_launch(void* const* d_in, const int* in_sizes, int n_in,
                              void* d_out, int out_size, void* d_ws, size_t ws_size,
                              hipStream_t stream) {
    const float* pred   = (const float*)d_in[0];
    const float* target = (const float*)d_in[1];
    const int*   lines  = (const int*)d_in[2];
    float* out = (float*)d_out;

    const int ntok = in_sizes[2];          // B * SEQ_LEN
    const int B    = ntok / SEQ_LEN;
    const float invN = 1.0f / (float)((long long)ntok * NCLS);

    safl_zero_kernel<<<1, 1, 0, stream>>>(out);
    safl_kernel<<<B * 4, BLOCK_T, 0, stream>>>(pred, target, lines, out, invN);
}